// AdjustedGATModel_22943715295837
// MI455X (gfx1250) — compile-verified
//
#include <hip/hip_runtime.h>
#include <cstddef>
#include <cstdint>

#define NNODES 10000
#define NEDGES 160000
#define FIN    512
#define DIM    256
#define HEADS  4
#define HID    1024
#define NLAYER 3
#define NGRAPH 16
#define NCLS   10
#define ETOT   (NEDGES + NNODES)

// GEMM tiling
#define BM     128            // rows per block
#define BN     128            // cols per block
#define TILEK  16             // K chunk staged in LDS (double buffered)
#define APITCH 20             // A LDS pitch (floats): 20m mod 64 -> 16 distinct banks
#define BPITCH 136            // B LDS pitch (floats): 2*136 % 64 = 16 -> halves disjoint

typedef __attribute__((ext_vector_type(2))) float v2f;
typedef __attribute__((ext_vector_type(8))) float v8f;

// ---------------------------------------------------------------------------
// CDNA5 async global->LDS copy (ASYNCcnt path), per ISA 15.18.3 opcode 98
// ---------------------------------------------------------------------------
__device__ inline void async_g2l_b128(uint32_t lds_off, uint64_t gaddr) {
    asm volatile("global_load_async_to_lds_b128 %0, %1, off"
                 :: "v"(lds_off), "v"(gaddr)
                 : "memory");
}
__device__ inline void wait_async0() {
    asm volatile("s_wait_asynccnt 0x0" ::: "memory");
}

// ---------------------------------------------------------------------------
// Utility fill
// ---------------------------------------------------------------------------
__global__ void fill_kernel(float* __restrict__ p, float v, int n) {
    int i = blockIdx.x * blockDim.x + threadIdx.x;
    if (i < n) p[i] = v;
}

// ---------------------------------------------------------------------------
// GEMM: C[M,1024] = A[M,K] @ B[K,1024] with V_WMMA_F32_16X16X4_F32.
// Block = 128x128 tile, 8 wave32: wave w -> rows [w*16,w*16+16), all 128 cols
// (8 accumulators). B tile (TILEK x 128) in LDS shared by all waves; A stripes
// (128 x TILEK) per wave. Double-buffered LDS: chunk i+1 staged by the async
// DMA path while chunk i feeds the WMMA pipe; one barrier per chunk.
// ---------------------------------------------------------------------------
__global__ __launch_bounds__(256) void gemm_wmma_f32(
    const float* __restrict__ A, const float* __restrict__ B,
    float* __restrict__ C, int K, int Nout, int M) {
    __shared__ float Als[2][BM * APITCH];       // 2 x 10 KB
    __shared__ float Bls[2][TILEK * BPITCH];    // 2 x 8.5 KB

    const int tid  = threadIdx.x;
    const int wave = tid >> 5;
    const int lane = tid & 31;
    const int half = lane >> 4;                 // 0: lanes 0-15, 1: lanes 16-31
    const int m    = lane & 15;
    const long rowBase = (long)blockIdx.x * BM;
    const int  cBase   = blockIdx.y * BN;

    v8f acc[8] = {};

    // ---- stage one K-chunk (kc) into buffer bi via async B128 copies ------
    auto stage = [&](int kc, int bi) {
        #pragma unroll
        for (int p = 0; p < 2; ++p) {
            int idx = tid + p * 256;            // 512 float4 per matrix tile
            // A: 128 rows x 4 float4
            int ar4 = idx >> 2, ac4 = idx & 3;
            long arow = rowBase + ar4;
            if (arow >= M) arow = M - 1;        // clamp: rows >= M never stored
            async_g2l_b128(
                (uint32_t)(uintptr_t)(&Als[bi][ar4 * APITCH + ac4 * 4]),
                (uint64_t)(uintptr_t)(A + arow * K + kc + ac4 * 4));
            // B: 16 rows x 32 float4
            int br = idx >> 5, bc4 = idx & 31;
            async_g2l_b128(
                (uint32_t)(uintptr_t)(&Bls[bi][br * BPITCH + bc4 * 4]),
                (uint64_t)(uintptr_t)(B + (long)(kc + br) * Nout + cBase +
                                      bc4 * 4));
        }
    };

    const int nchunks = K / TILEK;
    stage(0, 0);
    wait_async0();
    __syncthreads();

    for (int c = 0; c < nchunks; ++c) {
        const int cur = c & 1;
        if (c + 1 < nchunks) stage((c + 1) * TILEK, cur ^ 1);

        // ---- compute: 4 k-steps of WMMA f32 16x16x4 over 8 col tiles ------
        const float* __restrict__ Ab = &Als[cur][(wave * 16 + m) * APITCH];
        const float* __restrict__ Bb = &Bls[cur][0];
        #pragma unroll
        for (int k0 = 0; k0 < TILEK; k0 += 4) {
            // A frag: lanes0-15 K=k0,k0+1 ; lanes16-31 K=k0+2,k0+3
            v2f a;
            a.x = Ab[k0 + 2 * half];
            a.y = Ab[k0 + 2 * half + 1];
            const int bbase = (k0 + 2 * half) * BPITCH + m;
            #pragma unroll
            for (int t = 0; t < 8; ++t) {
                v2f b;                          // mirrored 4x16 layout
                b.x = Bb[bbase + t * 16];
                b.y = Bb[bbase + BPITCH + t * 16];
                acc[t] = __builtin_amdgcn_wmma_f32_16x16x4_f32(
                    false, a, false, b, (short)0, acc[t], false, false);
            }
        }

        wait_async0();       // next chunk's DMA done (overlapped with compute)
        __syncthreads();     // all waves done reading cur + next buffer ready
    }

    // ---- store: C/D layout VGPR i -> M = i (+8 upper half), N = lane&15 ---
    #pragma unroll
    for (int t = 0; t < 8; ++t) {
        int col = cBase + t * 16 + m;
        #pragma unroll
        for (int i = 0; i < 8; ++i) {
            long row = rowBase + wave * 16 + i + 8 * half;
            if (row < M) C[row * Nout + col] = acc[t][i];
        }
    }
}

// ---------------------------------------------------------------------------
// Per-node attention scores: e_src[n,h] = dot(h[n,h,:], a_src[h,:]) (and dst)
// ---------------------------------------------------------------------------
__global__ __launch_bounds__(256) void node_scores_kernel(
    const float* __restrict__ h, const float* __restrict__ as,
    const float* __restrict__ ad, float* __restrict__ es, float* __restrict__ ed) {
    int wid  = (blockIdx.x * blockDim.x + threadIdx.x) >> 5;
    int lane = threadIdx.x & 31;
    if (wid >= NNODES * HEADS) return;
    int n = wid >> 2, hh = wid & 3;
    const float* hp = h + (size_t)n * HID + hh * DIM + lane * 8;
    const float* ap = as + hh * DIM + lane * 8;
    const float* bp = ad + hh * DIM + lane * 8;
    float ss = 0.f, sd = 0.f;
    #pragma unroll
    for (int j = 0; j < 8; ++j) { ss += hp[j] * ap[j]; sd += hp[j] * bp[j]; }
    #pragma unroll
    for (int off = 16; off; off >>= 1) {
        ss += __shfl_xor(ss, off, 32);
        sd += __shfl_xor(sd, off, 32);
    }
    if (lane == 0) { es[wid] = ss; ed[wid] = sd; }
}

// monotonic float atomic-max via int/uint encodings (emax initialized to -inf)
__device__ inline void atomicMaxF(float* addr, float v) {
    if (v >= 0.f) atomicMax((int*)addr, __float_as_int(v));
    else          atomicMin((unsigned int*)addr, (unsigned int)__float_as_int(v));
}

// ---------------------------------------------------------------------------
// Edge pass 1: leaky-relu score + segment max over dst
// ---------------------------------------------------------------------------
__global__ __launch_bounds__(256) void edge_max_kernel(
    const int* __restrict__ ei, const float* __restrict__ es,
    const float* __restrict__ ed, float* __restrict__ emax,
    float* __restrict__ ebuf) {
    int e = blockIdx.x * blockDim.x + threadIdx.x;
    if (e >= ETOT) return;
    int src = (e < NEDGES) ? ei[e] : (e - NEDGES);
    int dst = (e < NEDGES) ? ei[NEDGES + e] : (e - NEDGES);
    #pragma unroll
    for (int hh = 0; hh < HEADS; ++hh) {
        float v = es[src * HEADS + hh] + ed[dst * HEADS + hh];
        v = (v > 0.f) ? v : 0.2f * v;                 // leaky_relu(0.2)
        ebuf[(size_t)e * HEADS + hh] = v;
        atomicMaxF(&emax[dst * HEADS + hh], v);
    }
}

// ---------------------------------------------------------------------------
// Edge pass 2: exp(e - max) + segment sum
// ---------------------------------------------------------------------------
__global__ __launch_bounds__(256) void edge_exp_kernel(
    const int* __restrict__ ei, const float* __restrict__ emax,
    float* __restrict__ ebuf, float* __restrict__ denom) {
    int e = blockIdx.x * blockDim.x + threadIdx.x;
    if (e >= ETOT) return;
    int dst = (e < NEDGES) ? ei[NEDGES + e] : (e - NEDGES);
    #pragma unroll
    for (int hh = 0; hh < HEADS; ++hh) {
        float t = __expf(ebuf[(size_t)e * HEADS + hh] - emax[dst * HEADS + hh]);
        ebuf[(size_t)e * HEADS + hh] = t;
        atomicAdd(&denom[dst * HEADS + hh], t);
    }
}

// ---------------------------------------------------------------------------
// Edge pass 3: out[dst] += h[src] * alpha   (wave32 per edge, 8 f32/lane/head)
// h (40 MB) is resident in the 192 MB L2, so gather+atomics stay on-chip.
// ---------------------------------------------------------------------------
__global__ __launch_bounds__(256) void aggregate_kernel(
    const int* __restrict__ ei, const float* __restrict__ h,
    const float* __restrict__ ebuf, const float* __restrict__ denom,
    float* __restrict__ outagg) {
    int wid  = (blockIdx.x * blockDim.x + threadIdx.x) >> 5;
    int lane = threadIdx.x & 31;
    if (wid >= ETOT) return;
    int src = (wid < NEDGES) ? ei[wid] : (wid - NEDGES);
    int dst = (wid < NEDGES) ? ei[NEDGES + wid] : (wid - NEDGES);
    #pragma unroll
    for (int hh = 0; hh < HEADS; ++hh) {
        float alpha = ebuf[(size_t)wid * HEADS + hh] / denom[dst * HEADS + hh];
        const float* hs = h + (size_t)src * HID + hh * DIM + lane * 8;
        float* od = outagg + (size_t)dst * HID + hh * DIM + lane * 8;
        float4 v0 = *(const float4*)hs;
        float4 v1 = *(const float4*)(hs + 4);
        atomicAdd(od + 0, v0.x * alpha); atomicAdd(od + 1, v0.y * alpha);
        atomicAdd(od + 2, v0.z * alpha); atomicAdd(od + 3, v0.w * alpha);
        atomicAdd(od + 4, v1.x * alpha); atomicAdd(od + 5, v1.y * alpha);
        atomicAdd(od + 6, v1.z * alpha); atomicAdd(od + 7, v1.w * alpha);
    }
}

// ---------------------------------------------------------------------------
// bias + ReLU (in place on aggregation buffer)
// ---------------------------------------------------------------------------
__global__ void bias_relu_kernel(float* __restrict__ a, const float* __restrict__ b,
                                 int n) {
    int i = blockIdx.x * blockDim.x + threadIdx.x;
    if (i < n) a[i] = fmaxf(a[i] + b[i % HID], 0.f);
}

// ---------------------------------------------------------------------------
// Pooling: atomic segment sums over batch ids, plus counts
// ---------------------------------------------------------------------------
__global__ void pool_sum_kernel(const float* __restrict__ h,
                                const int* __restrict__ batch,
                                float* __restrict__ pooled) {
    int i = blockIdx.x * blockDim.x + threadIdx.x;
    if (i >= NNODES * HID) return;
    int n = i / HID, j = i % HID;
    atomicAdd(&pooled[batch[n] * HID + j], h[i]);
}

__global__ void count_kernel(const int* __restrict__ batch, float* __restrict__ cnt) {
    int n = blockIdx.x * blockDim.x + threadIdx.x;
    if (n < NNODES) atomicAdd(&cnt[batch[n]], 1.f);
}

// ---------------------------------------------------------------------------
// Mean + FC + log_softmax: one block per graph
// ---------------------------------------------------------------------------
__global__ __launch_bounds__(256) void fc_logsoftmax_kernel(
    const float* __restrict__ pooled, const float* __restrict__ cnt,
    const float* __restrict__ fcw, const float* __restrict__ fcb,
    float* __restrict__ out) {
    __shared__ float red[256];
    __shared__ float logits[NCLS];
    int g = blockIdx.x, t = threadIdx.x;
    float inv = 1.f / fmaxf(cnt[g], 1.f);
    for (int c = 0; c < NCLS; ++c) {
        float s = 0.f;
        for (int j = t; j < HID; j += 256)
            s += pooled[g * HID + j] * inv * fcw[j * NCLS + c];
        red[t] = s; __syncthreads();
        for (int off = 128; off; off >>= 1) {
            if (t < off) red[t] += red[t + off];
            __syncthreads();
        }
        if (t == 0) logits[c] = red[0] + fcb[c];
        __syncthreads();
    }
    if (t == 0) {
        float mx = -1e30f;
        for (int c = 0; c < NCLS; ++c) mx = fmaxf(mx, logits[c]);
        float se = 0.f;
        for (int c = 0; c < NCLS; ++c) se += __expf(logits[c] - mx);
        float lse = __logf(se);
        for (int c = 0; c < NCLS; ++c) out[g * NCLS + c] = logits[c] - mx - lse;
    }
}

// ---------------------------------------------------------------------------
// Host side
// ---------------------------------------------------------------------------
extern "C" void kernel_launch(void* const* d_in, const int* in_sizes, int n_in,
                              void* d_out, int out_size, void* d_ws, size_t ws_size,
                              hipStream_t stream) {
    (void)in_sizes; (void)n_in; (void)out_size; (void)ws_size;
    const float* x     = (const float*)d_in[0];
    const int*   ei    = (const int*)d_in[1];
    const int*   batch = (const int*)d_in[2];
    const float* W[3]  = {(const float*)d_in[3], (const float*)d_in[4],
                          (const float*)d_in[5]};
    const float* a_src = (const float*)d_in[6];   // [L,H,D]
    const float* a_dst = (const float*)d_in[7];
    const float* bias  = (const float*)d_in[8];   // [L,HID]
    const float* fcw   = (const float*)d_in[9];
    const float* fcb   = (const float*)d_in[10];
    float* out = (float*)d_out;

    // scratch layout
    float* p      = (float*)d_ws;
    float* hbuf   = p; p += (size_t)NNODES * HID;     // 40.96 MB
    float* aggbuf = p; p += (size_t)NNODES * HID;     // 40.96 MB
    float* esrc   = p; p += NNODES * HEADS;
    float* edst   = p; p += NNODES * HEADS;
    float* emax   = p; p += NNODES * HEADS;
    float* denom  = p; p += NNODES * HEADS;
    float* ebuf   = p; p += (size_t)ETOT * HEADS;
    float* pooled = p; p += NGRAPH * HID;
    float* cnt    = p; p += NGRAPH;

    const int NH = NNODES * HEADS;
    const float NEG_INF = -__builtin_huge_valf();
    const dim3 ggrid((NNODES + BM - 1) / BM, HID / BN);

    const float* in = x;
    int K = FIN;
    for (int l = 0; l < NLAYER; ++l) {
        // h = in @ W[l]
        gemm_wmma_f32<<<ggrid, 256, 0, stream>>>(in, W[l], hbuf, K, HID, NNODES);
        // attention score dots
        node_scores_kernel<<<(NH * 32 + 255) / 256, 256, 0, stream>>>(
            hbuf, a_src + (size_t)l * HEADS * DIM, a_dst + (size_t)l * HEADS * DIM,
            esrc, edst);
        // reset segment buffers
        fill_kernel<<<(NH + 255) / 256, 256, 0, stream>>>(emax, NEG_INF, NH);
        fill_kernel<<<(NH + 255) / 256, 256, 0, stream>>>(denom, 0.f, NH);
        fill_kernel<<<(NNODES * HID + 255) / 256, 256, 0, stream>>>(
            aggbuf, 0.f, NNODES * HID);
        // segment softmax
        edge_max_kernel<<<(ETOT + 255) / 256, 256, 0, stream>>>(ei, esrc, edst,
                                                                emax, ebuf);
        edge_exp_kernel<<<(ETOT + 255) / 256, 256, 0, stream>>>(ei, emax, ebuf,
                                                                denom);
        // weighted aggregation
        aggregate_kernel<<<(ETOT * 32 + 255) / 256, 256, 0, stream>>>(
            ei, hbuf, ebuf, denom, aggbuf);
        // bias + relu
        bias_relu_kernel<<<(NNODES * HID + 255) / 256, 256, 0, stream>>>(
            aggbuf, bias + (size_t)l * HID, NNODES * HID);
        in = aggbuf;
        K = HID;
    }

    // global mean pool + classifier
    fill_kernel<<<(NGRAPH * HID + NGRAPH + 255) / 256, 256, 0, stream>>>(
        pooled, 0.f, NGRAPH * HID + NGRAPH);   // cnt sits right after pooled
    pool_sum_kernel<<<(NNODES * HID + 255) / 256, 256, 0, stream>>>(aggbuf, batch,
                                                                    pooled);
    count_kernel<<<(NNODES + 255) / 256, 256, 0, stream>>>(batch, cnt);
    fc_logsoftmax_kernel<<<NGRAPH, 256, 0, stream>>>(pooled, cnt, fcw, fcb, out);
}